// QFF_72249939853322
// MI455X (gfx1250) — compile-verified
//
#include <hip/hip_runtime.h>
#include <math.h>

// QFF trilinear feature sampling for MI455X (gfx1250, wave32).
// One thread = one point. 16 freqs x 8 corners = 128 grids, 8 gathers each.
// cv (128MB) stays L2-resident; output streamed NT via async LDS->global.
// Per-axis trilinear setup hoisted: 6 axis descriptors per freq instead of 24.

#define NFREQ    16
#define GPF      8          // corners (grids per freq)
#define GRID_SZ  (64*64*64) // one 64^3 grid, floats
#define OUTC     131        // 3 coords + 128 features
#define TPB      256
#define ROWS     133        // padded LDS row stride in floats (stride*4 mod 256 -> conflict-free)

struct Axis {
    float w0, w1;   // boundary-masked lerp weights
    int   i0, i1;   // clamped indices, pre-multiplied by axis stride
};

__device__ __forceinline__ Axis axis_setup(float e, int stride) {
    // ((e+1)*64 - 1) * 0.5 = e*32 + 31.5, range [-0.5, 63.5]
    const float q  = fmaf(e, 32.0f, 31.5f);
    const float f0 = floorf(q);
    const float t  = q - f0;
    const int   i  = (int)f0;                 // in [-1, 63]
    Axis a;
    a.w0 = (i >= 0) ? (1.0f - t) : 0.0f;      // tap i   valid iff i >= 0
    a.w1 = (i < 63) ? t : 0.0f;               // tap i+1 valid iff i+1 <= 63
    a.i0 = max(i, 0) * stride;
    a.i1 = min(i + 1, 63) * stride;
    return a;
}

__global__ __launch_bounds__(TPB)
void qff_kernel(const float* __restrict__ points,
                const float* __restrict__ cv,
                const float* __restrict__ freqs,
                float* __restrict__ out,
                int npts)
{
    __shared__ float tile[TPB * ROWS];

    const int t = threadIdx.x;
    const int nbase = blockIdx.x * TPB;
    const int n = nbase + t;
    const bool alive = (n < npts);

    float px = 0.f, py = 0.f, pz = 0.f;
    if (alive) {
        px = points[3 * n + 0];
        py = points[3 * n + 1];
        pz = points[3 * n + 2];
    }

    float* row = &tile[t * ROWS];
    row[0] = px; row[1] = py; row[2] = pz;

    #pragma unroll 1
    for (int f = 0; f < NFREQ; ++f) {
        const float fr = freqs[f];                 // uniform -> s_load
        // 6 axis descriptors: [0]=sin variant, [1]=cos variant
        Axis ax[2], ay[2], az[2];
        ax[0] = axis_setup(__sinf(fr * px), 1);
        ax[1] = axis_setup(__cosf(fr * px), 1);
        ay[0] = axis_setup(__sinf(fr * py), 64);
        ay[1] = axis_setup(__cosf(fr * py), 64);
        az[0] = axis_setup(__sinf(fr * pz), 4096);
        az[1] = axis_setup(__cosf(fr * pz), 4096);

        const float* __restrict__ gbase = cv + (size_t)f * GPF * GRID_SZ;

        #pragma unroll
        for (int c = 0; c < GPF; ++c) {
            // corner bit2 -> x uses cos, bit1 -> y, bit0 -> z (compile-time select)
            const Axis X = ax[(c >> 2) & 1];
            const Axis Y = ay[(c >> 1) & 1];
            const Axis Z = az[c & 1];

            const float* __restrict__ g = gbase + c * GRID_SZ;
            const int r00 = Z.i0 + Y.i0, r01 = Z.i0 + Y.i1;
            const int r10 = Z.i1 + Y.i0, r11 = Z.i1 + Y.i1;
            // 8 L2-resident gathers
            const float v000 = g[r00 + X.i0], v001 = g[r00 + X.i1];
            const float v010 = g[r01 + X.i0], v011 = g[r01 + X.i1];
            const float v100 = g[r10 + X.i0], v101 = g[r10 + X.i1];
            const float v110 = g[r11 + X.i0], v111 = g[r11 + X.i1];

            const float a0 = Y.w0 * fmaf(X.w0, v000, X.w1 * v001)
                           + Y.w1 * fmaf(X.w0, v010, X.w1 * v011);
            const float a1 = Y.w0 * fmaf(X.w0, v100, X.w1 * v101)
                           + Y.w1 * fmaf(X.w0, v110, X.w1 * v111);
            const float acc = fmaf(Z.w0, a0, Z.w1 * a1);

            row[3 + f * GPF + c] = acc * 10.0f;
        }
    }

    __syncthreads();   // tile complete; flush reads other threads' rows

    // Coalesced non-temporal flush: async LDS -> global (ASYNCcnt path).
    // Block's output region is contiguous: [nbase*131, (nbase+rows)*131).
    const int nrows = min(TPB, npts - nbase);
    const int limit = nrows * OUTC;
    const float* outBase = out + (size_t)nbase * OUTC;

    #pragma unroll 1
    for (int k = 0; k < OUTC; ++k) {
        const int idx = k * TPB + t;
        if (idx < limit) {
            const int r   = idx / OUTC;            // magic-mul division
            const int col = idx - r * OUTC;
            // addrspace(3) offset: low 32 bits of flat LDS pointer
            const unsigned lds_off = (unsigned)(size_t)&tile[r * ROWS + col];
            const unsigned goff    = (unsigned)idx * 4u;
            asm volatile(
                "global_store_async_from_lds_b32 %0, %1, %2 th:TH_STORE_NT\n"
                :
                : "v"(goff), "v"(lds_off), "s"(outBase)
                : "memory");
        }
    }
    asm volatile("s_wait_asynccnt 0" ::: "memory");
}

extern "C" void kernel_launch(void* const* d_in, const int* in_sizes, int n_in,
                              void* d_out, int out_size, void* d_ws, size_t ws_size,
                              hipStream_t stream) {
    const float* points = (const float*)d_in[0];   // (N,3) f32
    const float* cv     = (const float*)d_in[1];   // (128,1,64,64,64) f32
    const float* freqs  = (const float*)d_in[2];   // (16,) f32
    float* out = (float*)d_out;                    // (N,131) f32

    const int npts = in_sizes[0] / 3;
    const int blocks = (npts + TPB - 1) / TPB;
    qff_kernel<<<blocks, TPB, 0, stream>>>(points, cv, freqs, out, npts);
}